// monomer_split_79800492359956
// MI455X (gfx1250) — compile-verified
//
#include <hip/hip_runtime.h>
#include <hip/hip_bf16.h>
#include <stdint.h>

// ---------------------------------------------------------------------------
// monomer_split for MI455X (gfx1250).
// Pure data-movement workload: ~566 MB written, ~0.3-0.5 GB read => HBM-bound,
// floor ~45us @ 23.3 TB/s. No matmul => no WMMA; the CDNA5 features that
// apply are the async global<->LDS DMA path (ASYNCcnt) + global prefetch.
// E1/E2 (512 MB) ride a depth-2 async pipeline with a 4-deep LDS ring buffer.
// ---------------------------------------------------------------------------

#define AS1 __attribute__((address_space(1)))
#define AS3 __attribute__((address_space(3)))

typedef int v4i __attribute__((vector_size(16)));

constexpr int B  = 16;
constexpr int L  = 256;
constexpr int H  = 256;
constexpr int EH = 64;    // floats per E row slice (256 bytes = 16 x b128)
constexpr int P  = 256;   // 2*M == 2*N

#if defined(__HIP_DEVICE_COMPILE__) && \
    __has_builtin(__builtin_amdgcn_global_load_async_to_lds_b128) && \
    __has_builtin(__builtin_amdgcn_global_store_async_from_lds_b128)
#define ASYNC_PATH 1
#else
#define ASYNC_PATH 0
#endif

template <int N>
__device__ __forceinline__ void wait_asyncN() {
#if defined(__HIP_DEVICE_COMPILE__)
#if __has_builtin(__builtin_amdgcn_s_wait_asynccnt)
  __builtin_amdgcn_s_wait_asynccnt(N);
#else
  asm volatile("s_wait_asynccnt %0" ::"i"(N) : "memory");
#endif
#endif
}

// ---------------------------------------------------------------------------
// E1/E2 gather: out[bb,i,j,:] = blk(i,j) ? E[bb,src(i),src(j),:] : 0
// One block per (bb,i). 8 waves; each wave moves 2 gathered 256B rows per
// round through a per-wave 4-deep LDS ring buffer using async DMA with two
// operations in flight (s_wait_asynccnt <= 1). Masked rows store from a
// zeroed LDS region (per-lane LDS source select).
// ---------------------------------------------------------------------------
__global__ __launch_bounds__(256) void gather_E_kernel(
    const float* __restrict__ Ein, float* __restrict__ Eout,
    const int* __restrict__ mol, int second)
{
  __shared__ float zbuf[128];        // 512 B of zeros (per-lane 16B slots)
  __shared__ float buf[8][4][128];   // per-wave 4-deep ring, 512 B per slot

  const int tid  = threadIdx.x;
  const int wv   = tid >> 5;
  const int lane = tid & 31;

  if (tid < 128) zbuf[tid] = 0.0f;

  const int bb = blockIdx.x >> 8;
  const int i  = blockIdx.x & 255;

  const int l0 = mol[2 * bb];
  const int l1 = mol[2 * bb + 1];
  const int la   = second ? l1 : l0;
  const int base = second ? l0 : 0;

  // row-uniform (scalar) values
  const bool fI = i < la;
  const bool sI = (i >= la) && (i < 2 * la);
  int srcI = base + (fI ? i : i - la);
  srcI = srcI > (L - 1) ? (L - 1) : srcI;

  __syncthreads();  // zbuf visible (DS writes drained) before async engine reads

  const size_t inRow  = (size_t)(bb * L + srcI) * (size_t)L;  // row-slice index
  const size_t outRow = (size_t)blockIdx.x * (size_t)P;

  const int jsub = lane >> 4;  // which of the 2 rows this wave handles
  const int q    = lane & 15;  // 16B chunk within the 256B row

  // Per-round j for this lane: j(r) = r*16 + wv*2 + jsub
  auto src_of = [&](int j) -> int {
    int s = base + ((j < la) ? j : j - la);
    return s > (L - 1) ? (L - 1) : s;
  };
  auto keep_of = [&](int j) -> bool {
    const bool fJ = j < la;
    const bool sJ = (j >= la) && (j < 2 * la);
    return (fI && fJ) || (sI && sJ);
  };
  auto gsrc_of = [&](int j) -> const float* {
    return Ein + (inRow + (size_t)src_of(j)) * EH + q * 4;
  };

#if ASYNC_PATH
  // ---- depth-2 async pipeline over 16 rounds ----
  {
    const int j0 = wv * 2 + jsub;
    __builtin_amdgcn_global_load_async_to_lds_b128(
        (AS1 v4i*)gsrc_of(j0), (AS3 v4i*)&buf[wv][0][lane * 4], 0, 0);
  }
#pragma unroll
  for (int r = 0; r < 16; ++r) {
    const int j = r * 16 + wv * 2 + jsub;
    if (r < 15) {
      const int jn = j + 16;
      __builtin_prefetch((const void*)gsrc_of(jn), 0, 0);
      __builtin_amdgcn_global_load_async_to_lds_b128(
          (AS1 v4i*)gsrc_of(jn), (AS3 v4i*)&buf[wv][(r + 1) & 3][lane * 4], 0, 0);
      wait_asyncN<1>();   // load r complete (in-order load completion)
    } else {
      wait_asyncN<0>();   // no trailing load: drain everything
    }
    const float* lsrc = keep_of(j) ? &buf[wv][r & 3][lane * 4] : &zbuf[lane * 4];
    float* gdst = Eout + (outRow + (size_t)j) * EH + q * 4;
    __builtin_amdgcn_global_store_async_from_lds_b128(
        (AS1 v4i*)gdst, (AS3 v4i*)lsrc, 0, 0);
  }
  wait_asyncN<0>();
#else
#pragma unroll
  for (int r = 0; r < 16; ++r) {
    const int j = r * 16 + wv * 2 + jsub;
    float* gdst = Eout + (outRow + (size_t)j) * EH + q * 4;
    float4 v;
    if (keep_of(j)) v = *(const float4*)gsrc_of(j);
    else            v = make_float4(0.f, 0.f, 0.f, 0.f);
    *(float4*)gdst = v;
  }
#endif
}

// ---------------------------------------------------------------------------
// X1/X2: out[bb,p,c] = (p < 2*la) ? X[bb, src(p), c] : 0   (base always 0)
// ---------------------------------------------------------------------------
__global__ __launch_bounds__(256) void pack_rows_kernel(
    const float* __restrict__ X, float* __restrict__ out,
    const int* __restrict__ mol, int second)
{
  const int bb = blockIdx.x >> 8;
  const int p  = blockIdx.x & 255;
  const int c  = threadIdx.x;
  const int l0 = mol[2 * bb];
  const int l1 = mol[2 * bb + 1];
  const int la = second ? l1 : l0;
  const bool valid = p < 2 * la;
  int src = (p < la) ? p : p - la;
  src = src > (L - 1) ? (L - 1) : src;
  out[(size_t)blockIdx.x * H + c] =
      valid ? X[((size_t)bb * L + src) * H + c] : 0.0f;
}

// ---------------------------------------------------------------------------
// A1/A2 + mask1/mask2 (bools written as 0/1 f32)
// ---------------------------------------------------------------------------
__global__ __launch_bounds__(256) void pair_small_kernel(
    const float* __restrict__ A, float* __restrict__ outA,
    float* __restrict__ outM, const int* __restrict__ mol, int second)
{
  const int bb = blockIdx.x >> 8;
  const int i  = blockIdx.x & 255;
  const int j  = threadIdx.x;
  const int l0 = mol[2 * bb];
  const int l1 = mol[2 * bb + 1];
  const int la   = second ? l1 : l0;
  const int base = second ? l0 : 0;

  const bool fI = i < la, sI = (i >= la) && (i < 2 * la);
  const bool fJ = j < la, sJ = (j >= la) && (j < 2 * la);
  int srcI = base + (fI ? i : i - la); srcI = srcI > (L - 1) ? (L - 1) : srcI;
  int srcJ = base + (fJ ? j : j - la); srcJ = srcJ > (L - 1) ? (L - 1) : srcJ;

  const bool blk = (fI && fJ) || (sI && sJ);
  const size_t o = (size_t)blockIdx.x * P + j;
  outA[o] = blk ? A[((size_t)bb * L + srcI) * L + srcJ] : 0.0f;
  outM[o] = (!((fI && sJ) || (sI && fJ))) ? 1.0f : 0.0f;
}

// ---------------------------------------------------------------------------
// mask_out (bool -> 0/1 f32); mask input read as 1-byte bools
// ---------------------------------------------------------------------------
__global__ __launch_bounds__(256) void mask_out_kernel(
    const unsigned char* __restrict__ mask, float* __restrict__ out,
    const int* __restrict__ mol)
{
  const int bb = blockIdx.x >> 8;
  const int q  = blockIdx.x & 255;
  const int r  = threadIdx.x;
  const int l0 = mol[2 * bb];
  const int l1 = mol[2 * bb + 1];
  const int ls = l0 + l1;

  const bool lq = q < l0, lr = r < l0;
  const bool mq = (q >= l0) && (q < ls);
  const bool mr = (r >= l0) && (r < ls);
  const bool ct = (lq && lr) || (!lq && !lr);
  const bool cf = (lq && mr) || (mq && lr);

  const size_t o = ((size_t)bb * L + q) * L + r;
  float v;
  if (ct)      v = 1.0f;
  else if (cf) v = 0.0f;
  else         v = mask[o] ? 1.0f : 0.0f;
  out[o] = v;
}

__global__ __launch_bounds__(256) void zero_fill_kernel(float* __restrict__ out, int n)
{
  const int idx = blockIdx.x * blockDim.x + threadIdx.x;
  if (idx < n) out[idx] = 0.0f;
}

// ---------------------------------------------------------------------------
extern "C" void kernel_launch(void* const* d_in, const int* in_sizes, int n_in,
                              void* d_out, int out_size, void* d_ws, size_t ws_size,
                              hipStream_t stream) {
  const float* X = (const float*)d_in[0];
  const float* E = (const float*)d_in[1];
  const float* A = (const float*)d_in[2];
  // d_in[3] = w (unused: w1/w2 are zeros)
  const unsigned char* mask = (const unsigned char*)d_in[4];
  const int* mol = (const int*)d_in[5];

  float* out = (float*)d_out;
  float* oX1 = out;
  float* oX2 = oX1 + (size_t)B * P * H;
  float* oE1 = oX2 + (size_t)B * P * H;
  float* oE2 = oE1 + (size_t)B * P * P * EH;
  float* oA1 = oE2 + (size_t)B * P * P * EH;
  float* oA2 = oA1 + (size_t)B * P * P;
  float* oM1 = oA2 + (size_t)B * P * P;
  float* oM2 = oM1 + (size_t)B * P * P;
  float* oW1 = oM2 + (size_t)B * P * P;
  float* oW2 = oW1 + (size_t)B * P;
  float* oMO = oW2 + (size_t)B * P;

  const dim3 blk(256);
  gather_E_kernel  <<<dim3(B * P), blk, 0, stream>>>(E, oE1, mol, 0);
  gather_E_kernel  <<<dim3(B * P), blk, 0, stream>>>(E, oE2, mol, 1);
  pack_rows_kernel <<<dim3(B * P), blk, 0, stream>>>(X, oX1, mol, 0);
  pack_rows_kernel <<<dim3(B * P), blk, 0, stream>>>(X, oX2, mol, 1);
  pair_small_kernel<<<dim3(B * P), blk, 0, stream>>>(A, oA1, oM1, mol, 0);
  pair_small_kernel<<<dim3(B * P), blk, 0, stream>>>(A, oA2, oM2, mol, 1);
  zero_fill_kernel <<<dim3((2 * B * P + 255) / 256), blk, 0, stream>>>(oW1, 2 * B * P);
  mask_out_kernel  <<<dim3(B * L), blk, 0, stream>>>(mask, oMO, mol);
}